// GraphAttentionLayer_86835648790653
// MI455X (gfx1250) — compile-verified
//
#include <hip/hip_runtime.h>
#include <hip/hip_bf16.h>
#include <math.h>

// GAT layer, N=8192, F=D=128, H=2, fused single-pass-over-adj design.
// ws layout (floats): value[2*8192*128] | out_acc[2*8192*128] | sa1[2*8192] |
//                     sa2[2*8192] | l[2*8192] | w1[256] | w2[256] | smax2[2]
// total ~16.2 MB of workspace.

typedef float v2f __attribute__((ext_vector_type(2)));
typedef float v4f __attribute__((ext_vector_type(4)));
typedef float v8f __attribute__((ext_vector_type(8)));

#define NN 8192
#define FF 128
#define DD 128
#define ALPHA 0.3f

// ---------------- K1: w1[h,f] = sum_d Wmap[h,f,d]*a1_w[h,d] (likewise w2) ----
__global__ __launch_bounds__(256) void gat_prep_vec(
    const float* __restrict__ Wmap, const float* __restrict__ a1w,
    const float* __restrict__ a2w, float* __restrict__ w1, float* __restrict__ w2) {
  int t = threadIdx.x;            // t = h*128 + f
  int h = t >> 7;
  const float* wrow = Wmap + (size_t)t * DD;
  const float* a1 = a1w + h * DD;
  const float* a2 = a2w + h * DD;
  float s1 = 0.f, s2 = 0.f;
  for (int d = 0; d < DD; d += 4) {
    v4f wv = *(const v4f*)(wrow + d);
    v4f x1 = *(const v4f*)(a1 + d);
    v4f x2 = *(const v4f*)(a2 + d);
    s1 += wv[0]*x1[0] + wv[1]*x1[1] + wv[2]*x1[2] + wv[3]*x1[3];
    s2 += wv[0]*x2[0] + wv[1]*x2[1] + wv[2]*x2[2] + wv[3]*x2[3];
  }
  w1[t] = s1;
  w2[t] = s2;
}

// ---------------- K2: sa1[h,n] = x[n,:].w1[h,:] + a1_b[h]  (likewise sa2) ----
__global__ __launch_bounds__(256) void gat_sa(
    const float* __restrict__ x, const float* __restrict__ w1,
    const float* __restrict__ w2, const float* __restrict__ a1b,
    const float* __restrict__ a2b, float* __restrict__ sa1, float* __restrict__ sa2) {
  __shared__ float xs[64 * 129];   // stride 129 -> conflict-free column reads
  __shared__ float wv[512];
  int t = threadIdx.x;
  int rb = blockIdx.x * 64;
  wv[t]       = w1[t];
  wv[t + 256] = w2[t];
  {
    int row = t >> 2, q = t & 3;
    const float* src = x + (size_t)(rb + row) * FF + q * 32;
#pragma unroll
    for (int i = 0; i < 8; ++i) {
      v4f v = *(const v4f*)(src + i * 4);
      float* dst = &xs[row * 129 + q * 32 + i * 4];
      dst[0] = v[0]; dst[1] = v[1]; dst[2] = v[2]; dst[3] = v[3];
    }
  }
  __syncthreads();
  int row = t & 63, h = (t >> 6) & 1, which = t >> 7;
  const float* w  = &wv[which * 256 + h * 128];
  const float* xr = &xs[row * 129];
  float acc = 0.f;
#pragma unroll 8
  for (int f = 0; f < FF; ++f) acc += xr[f] * w[f];
  if (which == 0) sa1[h * NN + rb + row] = acc + a1b[h];
  else            sa2[h * NN + rb + row] = acc + a2b[h];
}

// ---------------- K2b: smax2[h] = max_j sa2[h,j]  (safe softmax bound) ------
__global__ __launch_bounds__(256) void gat_smax(
    const float* __restrict__ sa2, float* __restrict__ smax2) {
  __shared__ float red[256];
  int t = threadIdx.x, h = t >> 7, j = t & 127;
  float m = -1e30f;
  for (int k = 0; k < 64; ++k) m = fmaxf(m, sa2[h * NN + j + 128 * k]);
  red[t] = m;
  __syncthreads();
  for (int s = 64; s > 0; s >>= 1) {
    if ((t & 127) < s) red[t] = fmaxf(red[t], red[t + s]);
    __syncthreads();
  }
  if ((t & 127) == 0) smax2[h] = red[t];
}

// ---------------- K3: value[h] = x @ kernel[h]  (f32 WMMA 16x16x4) ----------
// WMMA f32 operand layout assumed per ISA 7.12.2 (NOT hw-verified):
//  A 16x4: lane<16 -> M=lane, {v0,v1}={K0,K1}; lane>=16 -> M=lane-16, {K2,K3}
//  B 4x16: lane<16 -> N=lane, {v0,v1}={K0,K1}; lane>=16 -> N=lane-16, {K2,K3}
//  C/D: vgpr k -> M=k (lanes 0-15) / M=k+8 (lanes 16-31), N=lane%16
__global__ __launch_bounds__(256) void gat_value(
    const float* __restrict__ x, const float* __restrict__ kern,
    float* __restrict__ value_ws) {
  int t = threadIdx.x, lane = t & 31, wave = t >> 5;
  int h  = blockIdx.x >> 7;
  int rb = (blockIdx.x & 127) * 64;
  int mt = wave >> 1;
  int dg = (wave & 1) * 4;
  int rowA = lane & 15;
  int jsel = (lane < 16) ? 0 : 2;
  v8f zero = {};
  v8f acc[4];
#pragma unroll
  for (int i = 0; i < 4; ++i) acc[i] = zero;
  const float* xrow = x + (size_t)(rb + mt * 16 + rowA) * FF;
#pragma unroll 4
  for (int k = 0; k < 32; ++k) {
    int f0 = k * 4 + jsel;
    v2f a = *(const v2f*)(xrow + f0);
#pragma unroll
    for (int di = 0; di < 4; ++di) {
      int dt = dg + di;
      const float* bp = kern + (size_t)(h * FF + f0) * DD + dt * 16 + rowA;
      v2f b; b[0] = bp[0]; b[1] = bp[DD];
      acc[di] = __builtin_amdgcn_wmma_f32_16x16x4_f32(
          false, a, false, b, (short)0, acc[di], false, false);
    }
  }
  int mbase = rb + mt * 16 + ((lane < 16) ? 0 : 8);
#pragma unroll
  for (int di = 0; di < 4; ++di) {
    int d = (dg + di) * 16 + rowA;
#pragma unroll
    for (int k = 0; k < 8; ++k)
      value_ws[((size_t)h * NN + mbase + k) * DD + d] = acc[di][k];
  }
}

// ---------------- K4: fused masked-score + softmax(unnormalized) + p@value --
// grid = 128 row-blocks x 4 j-splits. One pass over adj (NT loads).
// p = exp(leaky(adj*(sa1_i+sa2_j)) - m) with safe m => no rescale, no 2nd pass.
__global__ __launch_bounds__(256) void gat_attn(
    const float* __restrict__ adj, const float* __restrict__ sa1,
    const float* __restrict__ sa2, const float* __restrict__ smax2,
    const float* __restrict__ value_ws, float* __restrict__ out_ws,
    float* __restrict__ l_ws) {
  __shared__ float p_lds[2 * 64 * 36];  // [h][row][j] stride 36: aligned + low-conflict
  __shared__ float vt[2 * 128 * 34];    // [h][d][j] stride 34: b64-aligned B reads
  int t = threadIdx.x, lane = t & 31, wave = t >> 5;
  int rb = (blockIdx.x >> 2) * 64;
  int jb = (blockIdx.x & 3) * 2048;

  // per-thread score/softmax assignment: rows rloc0, rloc0+32; 4 j's at j0
  int rloc0 = t >> 3, j0 = (t & 7) << 2;
  float sa1r[2][2], m_p[2][2];
#pragma unroll
  for (int h = 0; h < 2; ++h) {
    float sm = smax2[h];
#pragma unroll
    for (int pass = 0; pass < 2; ++pass) {
      float v = sa1[h * NN + rb + rloc0 + 32 * pass];
      sa1r[h][pass] = v;
      m_p[h][pass]  = fmaxf(0.f, v + sm);   // safe upper bound on row max
    }
  }
  float l_acc[2][2] = {{0.f, 0.f}, {0.f, 0.f}};

  // value-staging assignment
  int hv = t >> 7, u = t & 127, jv = u >> 2, db = (u & 3) << 5;
  // WMMA assignment: wave -> (mtile, head), owns all 8 d-tiles
  int mt = wave & 3, hw = wave >> 2;
  int rowA = lane & 15;
  int jsel = (lane < 16) ? 0 : 2;
  v8f zero = {};
  v8f acc[8];
#pragma unroll
  for (int i = 0; i < 8; ++i) acc[i] = zero;
  const float* prow = &p_lds[(hw * 64 + mt * 16 + rowA) * 36];

  for (int jt = 0; jt < 2048; jt += 32) {
    __syncthreads();
    {   // stage value tile transposed: vt[h][d][j]
      const float* src = value_ws + ((size_t)hv * NN + jb + jt + jv) * DD + db;
#pragma unroll
      for (int i = 0; i < 8; ++i) {
        v4f v = *(const v4f*)(src + i * 4);
#pragma unroll
        for (int c = 0; c < 4; ++c)
          vt[(hv * 128 + db + i * 4 + c) * 34 + jv] = v[c];
      }
    }
    {   // compute dense p tile (includes adj==0 entries: p = exp(-m))
      v4f s2v[2];
#pragma unroll
      for (int h = 0; h < 2; ++h)
        s2v[h] = *(const v4f*)(sa2 + h * NN + jb + jt + j0);
#pragma unroll
      for (int pass = 0; pass < 2; ++pass) {
        int row = rloc0 + 32 * pass;
        v4f av = __builtin_nontemporal_load(
            (const v4f*)(adj + (size_t)(rb + row) * NN + jb + jt + j0));
#pragma unroll
        for (int h = 0; h < 2; ++h) {
          float base = sa1r[h][pass], m = m_p[h][pass];
          v4f pv; float lsum = 0.f;
#pragma unroll
          for (int c = 0; c < 4; ++c) {
            float s = av[c] * (base + s2v[h][c]);   // adj in {0,1}
            float w = fmaxf(s, ALPHA * s);          // LeakyReLU(0.3)
            float p = __expf(w - m);
            pv[c] = p; lsum += p;
          }
          *(v4f*)&p_lds[(h * 64 + row) * 36 + j0] = pv;
          l_acc[h][pass] += lsum;
        }
      }
    }
    __syncthreads();
    // acc += p_tile @ value_tile via f32 WMMA, K=32 in 8 steps of 4
#pragma unroll
    for (int k = 0; k < 8; ++k) {
      v2f a = *(const v2f*)(prow + k * 4 + jsel);
#pragma unroll
      for (int dt = 0; dt < 8; ++dt) {
        v2f b = *(const v2f*)&vt[(hw * 128 + dt * 16 + rowA) * 34 + k * 4 + jsel];
        acc[dt] = __builtin_amdgcn_wmma_f32_16x16x4_f32(
            false, a, false, b, (short)0, acc[dt], false, false);
      }
    }
  }
  // combine j-split partials
  int mbase = rb + mt * 16 + ((lane < 16) ? 0 : 8);
#pragma unroll
  for (int dt = 0; dt < 8; ++dt) {
    int d = dt * 16 + rowA;
#pragma unroll
    for (int k = 0; k < 8; ++k)
      atomicAdd(&out_ws[((size_t)hw * NN + mbase + k) * DD + d], acc[dt][k]);
  }
#pragma unroll
  for (int h = 0; h < 2; ++h)
#pragma unroll
    for (int pass = 0; pass < 2; ++pass)
      atomicAdd(&l_ws[h * NN + rb + rloc0 + 32 * pass], l_acc[h][pass]);
}

// ---------------- K5: out = mean_h(acc_h / l_h + bias_h) --------------------
__global__ __launch_bounds__(256) void gat_final(
    const float* __restrict__ out_ws, const float* __restrict__ l_ws,
    const float* __restrict__ bias, float* __restrict__ out) {
  int idx = blockIdx.x * 256 + threadIdx.x;
  if (idx >= NN * DD) return;
  int n = idx >> 7;
  float v = 0.f;
#pragma unroll
  for (int h = 0; h < 2; ++h)
    v += out_ws[(size_t)h * (NN * DD) + idx] / l_ws[h * NN + n]
       + bias[(size_t)h * (NN * DD) + idx];
  out[idx] = 0.5f * v;
}

extern "C" void kernel_launch(void* const* d_in, const int* in_sizes, int n_in,
                              void* d_out, int out_size, void* d_ws, size_t ws_size,
                              hipStream_t stream) {
  (void)in_sizes; (void)n_in; (void)out_size; (void)ws_size;
  const float* x    = (const float*)d_in[0];
  const float* adj  = (const float*)d_in[1];
  const float* Wmap = (const float*)d_in[2];
  const float* a1w  = (const float*)d_in[3];
  const float* a1b  = (const float*)d_in[4];
  const float* a2w  = (const float*)d_in[5];
  const float* a2b  = (const float*)d_in[6];
  const float* kern = (const float*)d_in[7];
  const float* bias = (const float*)d_in[8];
  float* out = (float*)d_out;

  float* ws       = (float*)d_ws;
  float* value_ws = ws;                 // 2*8192*128
  float* out_ws   = ws + 2097152;       // 2*8192*128
  float* sa1      = ws + 4194304;       // 2*8192
  float* sa2      = ws + 4210688;       // 2*8192
  float* l_ws     = ws + 4227072;       // 2*8192
  float* w1       = ws + 4243456;       // 256
  float* w2       = ws + 4243712;       // 256
  float* smax2    = ws + 4243968;       // 2

  gat_prep_vec<<<1, 256, 0, stream>>>(Wmap, a1w, a2w, w1, w2);
  gat_sa<<<128, 256, 0, stream>>>(x, w1, w2, a1b, a2b, sa1, sa2);
  gat_smax<<<1, 256, 0, stream>>>(sa2, smax2);
  gat_value<<<256, 256, 0, stream>>>(x, kern, value_ws);
  hipMemsetAsync(out_ws, 0, 2097152 * sizeof(float), stream);
  hipMemsetAsync(l_ws, 0, 16384 * sizeof(float), stream);
  gat_attn<<<512, 256, 0, stream>>>(adj, sa1, sa2, smax2, value_ws, out_ws, l_ws);
  gat_final<<<4096, 256, 0, stream>>>(out_ws, l_ws, bias, out);
}